// Query_Attn_InteractMatrix_Network_18356690223531
// MI455X (gfx1250) — compile-verified
//
#include <hip/hip_runtime.h>
#include <math.h>

typedef __attribute__((ext_vector_type(2))) float v2f;
typedef __attribute__((ext_vector_type(4))) float v4f;
typedef __attribute__((ext_vector_type(8))) float v8f;

#define EMB 768
#define NB 32           // batch
#define NO 96           // number of W1 "heads"
#define NH 48           // layer-2 width
#define NT 48           // N-tiles of 16 covering EMB
#define XSTRIDE (3 * EMB)

// ---------------------------------------------------------------------------
// Kernel 1: U[o, 0:32, jb:jb+16] = Xq(32x768) @ W1[o](768x768) tile via
// V_WMMA_F32_16X16X4_F32, fused with the per-branch dot against Xp1/Xp2 over
// the 16 j's of this tile. One wave per (o, n-tile). Two 16x16 accumulators
// cover M=0..31 so each W1 element is loaded exactly once.
// Tile partials go to workspace: P1[(o*NT+nt)*NB + b], P2[...], summed in k2.
// ---------------------------------------------------------------------------
__global__ __launch_bounds__(32)
void interact_gemm_kernel(const float* __restrict__ X,
                          const float* __restrict__ W1,
                          float* __restrict__ P1,
                          float* __restrict__ P2)
{
    __shared__ float U[16 * 32];           // U[j][b] : 2 KB

    const int o    = blockIdx.x / NT;
    const int nt   = blockIdx.x % NT;
    const int jb   = nt * 16;
    const int lane = threadIdx.x;
    const int nl   = lane & 15;            // N-lane within tile; also M for A-frag
    const int half = lane >> 4;            // 0: lanes 0-15, 1: lanes 16-31
    const int koff = half << 1;            // A/B K sub-offset: 0 or 2

    // A-fragment rows: Xq row 'nl' (for c0) and row 'nl+16' (for c1)
    const float* xq0 = X + (size_t)nl * XSTRIDE;
    const float* xq1 = X + (size_t)(nl + 16) * XSTRIDE;
    // B-fragment column base: W1[o][*][jb + nl]
    const float* w = W1 + (size_t)o * EMB * EMB + jb + nl;

    v8f c0 = {};   // rows M = 0..15
    v8f c1 = {};   // rows M = 16..31

    #pragma unroll 4
    for (int k = 0; k < EMB; k += 4) {
        const int kk = k + koff;
        // A 16x4 frag: VGPR0 = K=koff, VGPR1 = K=koff+1 (contiguous -> b64 load)
        v2f a0 = { xq0[kk], xq0[kk + 1] };
        v2f a1 = { xq1[kk], xq1[kk + 1] };
        // B 4x16 frag: same K striping, N = nl; rows are EMB apart
        v2f b  = { w[(size_t)kk * EMB], w[(size_t)(kk + 1) * EMB] };
        c0 = __builtin_amdgcn_wmma_f32_16x16x4_f32(false, a0, false, b,
                                                   (short)0, c0, false, false);
        c1 = __builtin_amdgcn_wmma_f32_16x16x4_f32(false, a1, false, b,
                                                   (short)0, c1, false, false);
    }

    // --- Epilogue: dump U tile to LDS (transposed U[j][b]), then each lane
    // takes one batch row and dots it against Xp1/Xp2 of this j-tile. -------
    // D layout: reg r, lanes 0-15 -> (M=r, N=lane); lanes 16-31 -> (M=r+8).
    // c0 regs 0..7 therefore cover b = 8*half + 0..7  (contiguous!),
    // c1 regs 0..7 cover b = 16 + 8*half + 0..7.
    {
        const int ubase = nl * 32 + (half << 3);   // 32B-aligned
        *(v8f*)&U[ubase]      = c0;                // 2x ds_store_b128
        *(v8f*)&U[ubase + 16] = c1;                // 2x ds_store_b128
    }
    // Single wave per block: DS store->load ordering handled by s_wait_dscnt.

    const int b = lane;                            // batch row handled by lane
    const v4f* xp1v = (const v4f*)(X + (size_t)b * XSTRIDE + EMB + jb);
    const v4f* xp2v = (const v4f*)(X + (size_t)b * XSTRIDE + 2 * EMB + jb);

    v4f s1v = {}; v4f s2v = {};
    #pragma unroll
    for (int q = 0; q < 4; ++q) {
        const v4f x1 = xp1v[q];
        const v4f x2 = xp2v[q];
        v4f uq = { U[(q * 4 + 0) * 32 + b], U[(q * 4 + 1) * 32 + b],
                   U[(q * 4 + 2) * 32 + b], U[(q * 4 + 3) * 32 + b] };
        s1v += uq * x1;
        s2v += uq * x2;
    }
    const float s1 = (s1v[0] + s1v[1]) + (s1v[2] + s1v[3]);
    const float s2 = (s2v[0] + s2v[1]) + (s2v[2] + s2v[3]);

    const size_t base = (size_t)(o * NT + nt) * NB;
    P1[base + b] = s1;                             // fully coalesced
    P2[base + b] = s2;
}

// ---------------------------------------------------------------------------
// Kernel 2: per batch row b — sum tile partials over nt, +b1, ReLU,
// 96 -> 48 layer (+b2, ReLU), cosine similarity.
// ---------------------------------------------------------------------------
__global__ __launch_bounds__(96)
void head_kernel(const float* __restrict__ P1,
                 const float* __restrict__ P2,
                 const float* __restrict__ b1,
                 const float* __restrict__ W2,
                 const float* __restrict__ b2,
                 float* __restrict__ out)
{
    __shared__ float h1[NO];
    __shared__ float h2[NO];
    __shared__ float red[3][NH];

    const int b = blockIdx.x;
    const int t = threadIdx.x;

    if (t < NO) {
        float a1 = 0.f, a2 = 0.f;
        const size_t ob = (size_t)t * NT * NB + b;
        #pragma unroll
        for (int nt = 0; nt < NT; ++nt) {
            a1 += P1[ob + (size_t)nt * NB];
            a2 += P2[ob + (size_t)nt * NB];
        }
        h1[t] = fmaxf(a1 + b1[t], 0.f);
        h2[t] = fmaxf(a2 + b1[t], 0.f);
    }
    __syncthreads();

    if (t < NH) {
        float s1 = b2[t], s2 = b2[t];
        #pragma unroll
        for (int o = 0; o < NO; ++o) {
            const float w = W2[t * NO + o];
            s1 += w * h1[o];
            s2 += w * h2[o];
        }
        s1 = fmaxf(s1, 0.f);
        s2 = fmaxf(s2, 0.f);
        red[0][t] = s1 * s2;
        red[1][t] = s1 * s1;
        red[2][t] = s2 * s2;
    }
    __syncthreads();

    if (t == 0) {
        float num = 0.f, n1 = 0.f, n2 = 0.f;
        #pragma unroll
        for (int i = 0; i < NH; ++i) {
            num += red[0][i];
            n1  += red[1][i];
            n2  += red[2][i];
        }
        out[b] = num / fmaxf(sqrtf(n1) * sqrtf(n2), 1e-8f);
    }
}

// ---------------------------------------------------------------------------
extern "C" void kernel_launch(void* const* d_in, const int* in_sizes, int n_in,
                              void* d_out, int out_size, void* d_ws, size_t ws_size,
                              hipStream_t stream)
{
    const float* X  = (const float*)d_in[0];   // (32, 2304)
    const float* W1 = (const float*)d_in[1];   // (96, 768, 768)
    const float* b1 = (const float*)d_in[2];   // (96,)
    const float* W2 = (const float*)d_in[3];   // (48, 96)
    const float* b2 = (const float*)d_in[4];   // (48,)

    float* P1 = (float*)d_ws;                  // NO*NT*NB floats
    float* P2 = P1 + (size_t)NO * NT * NB;     // NO*NT*NB floats

    interact_gemm_kernel<<<NO * NT, 32, 0, stream>>>(X, W1, P1, P2);
    head_kernel<<<NB, 96, 0, stream>>>(P1, P2, b1, W2, b2, (float*)d_out);
}